// fuse_17325898072713
// MI455X (gfx1250) — compile-verified
//
#include <hip/hip_runtime.h>
#include <hip/hip_bf16.h>
#include <math.h>

typedef __attribute__((ext_vector_type(2))) float v2f;
typedef __attribute__((ext_vector_type(8))) float v8f;

#define LX 512
#define LY 128
#define BB 64
#define DD 512

// ---------------------------------------------------------------------------
// Kernel 1: per-row inverse norms for X [512*64 rows] and Y [128*64 rows].
// One wave per row of 512 floats. Linear row id r == x*64+b (matches [L,B,D]).
// ---------------------------------------------------------------------------
__global__ __launch_bounds__(256) void norm_kernel(const float* __restrict__ X,
                                                   const float* __restrict__ Y,
                                                   float* __restrict__ rnx,
                                                   float* __restrict__ rny) {
    const int gwave = (blockIdx.x * 256 + threadIdx.x) >> 5;  // 0..40959
    const int lane  = threadIdx.x & 31;
    const float* row;
    float* dst;
    if (gwave < LX * BB) { row = X + (size_t)gwave * DD;              dst = rnx + gwave; }
    else                 { int r = gwave - LX * BB;
                           row = Y + (size_t)r * DD;                  dst = rny + r; }
    float s = 0.0f;
#pragma unroll
    for (int j = 0; j < 4; ++j) {
        const float4 v = *(const float4*)(row + (j * 32 + lane) * 4);
        s += v.x * v.x + v.y * v.y + v.z * v.z + v.w * v.w;
    }
    s += __shfl_xor(s, 1, 32);
    s += __shfl_xor(s, 2, 32);
    s += __shfl_xor(s, 4, 32);
    s += __shfl_xor(s, 8, 32);
    s += __shfl_xor(s, 16, 32);
    if (lane == 0) *dst = 1.0f / fmaxf(sqrtf(s), 1e-12f);
}

// ---------------------------------------------------------------------------
// Kernel 2: WMMA batched GEMM (raw dot products, loads + WMMA only in the hot
// loop) + post-hoc normalization folded into the row-max / col-max epilogue.
// One wave = one 16-row x-tile for one b, spanning all 128 y (8 acc tiles).
// grid = 64 b * 4 x-groups blocks of 256 threads (8 waves).
// ---------------------------------------------------------------------------
__global__ __launch_bounds__(256) void gemm_hd_kernel(const float* __restrict__ X,
                                                      const float* __restrict__ Y,
                                                      const float* __restrict__ rnx,
                                                      const float* __restrict__ rny,
                                                      float* __restrict__ rowmin_partial,
                                                      float* __restrict__ colmax_partial) {
    const int lane  = threadIdx.x & 31;
    const int wave  = threadIdx.x >> 5;
    const int b     = blockIdx.x >> 2;
    const int xg    = blockIdx.x & 3;
    const int xTile = xg * 8 + wave;      // 0..31
    const int x0    = xTile * 16;
    const int m     = lane & 15;          // A row / B col within tile
    const int khalf = lane >> 4;          // 0 or 1 -> K offset 0 or 2

    // per-lane inverse norms (positive): lane j holds rnx[x0 + (j&15)]
    const float rnxl = rnx[(x0 + m) * BB + b];
    float rnyl[8];
#pragma unroll
    for (int t = 0; t < 8; ++t) rnyl[t] = rny[(t * 16 + m) * BB + b];

    const float* xrow = X + ((size_t)(x0 + m) * BB + b) * DD + 2 * khalf;
    const float* ybase[8];
#pragma unroll
    for (int t = 0; t < 8; ++t)
        ybase[t] = Y + ((size_t)(t * 16 + m) * BB + b) * DD + 2 * khalf;

    v8f acc[8];
#pragma unroll
    for (int t = 0; t < 8; ++t) acc[t] = (v8f){0, 0, 0, 0, 0, 0, 0, 0};

    // hot loop: pure loads + WMMA (normalization deferred to epilogue)
    for (int k = 0; k < DD; k += 4) {
        const v2f a = *(const v2f*)(xrow + k);
#pragma unroll
        for (int t = 0; t < 8; ++t) {
            const v2f bv = *(const v2f*)(ybase[t] + k);
            acc[t] = __builtin_amdgcn_wmma_f32_16x16x4_f32(
                false, a, false, bv, (short)0, acc[t], false, false);
        }
    }

    // C/D layout: vgpr i, lanes 0-15 -> M=i, lanes 16-31 -> M=8+i; N = lane&15.
    // Broadcast rnx scale per accumulator register: vgpr i needs rnx[x0+i+8*khalf].
    float sx[8];
#pragma unroll
    for (int i = 0; i < 8; ++i) sx[i] = __shfl(rnxl, i + 8 * khalf, 32);

    // rowmax over all y (all t, all N) per M, then min over this tile's 16 M.
    // rny factor varies per t -> apply before t-max; rnx factor constant per
    // (vgpr, half) -> apply after (all factors positive, commute with max).
    float rowmin = 1e30f;
#pragma unroll
    for (int i = 0; i < 8; ++i) {
        float v = acc[0][i] * rnyl[0];
#pragma unroll
        for (int t = 1; t < 8; ++t) v = fmaxf(v, acc[t][i] * rnyl[t]);
        v *= sx[i];
        v = fmaxf(v, __shfl_xor(v, 1, 32));
        v = fmaxf(v, __shfl_xor(v, 2, 32));
        v = fmaxf(v, __shfl_xor(v, 4, 32));
        v = fmaxf(v, __shfl_xor(v, 8, 32));
        rowmin = fminf(rowmin, v);
    }
    rowmin = fminf(rowmin, __shfl_xor(rowmin, 16, 32));
    if (lane == 0) rowmin_partial[b * 32 + xTile] = rowmin;

    // colmax over this tile's 16 M for each y; rnx factor varies per vgpr ->
    // apply before i-max; rny factor constant per lane -> apply after.
#pragma unroll
    for (int t = 0; t < 8; ++t) {
        float cm = acc[t][0] * sx[0];
#pragma unroll
        for (int i = 1; i < 8; ++i) cm = fmaxf(cm, acc[t][i] * sx[i]);
        cm *= rnyl[t];
        cm = fmaxf(cm, __shfl_xor(cm, 16, 32));
        if (lane < 16)
            colmax_partial[((size_t)(b * 32 + xTile)) * LY + t * 16 + lane] = cm;
    }
}

// ---------------------------------------------------------------------------
// Kernel 3: per-b reduction -> hd[b]. 64 blocks of 128 threads.
// ---------------------------------------------------------------------------
__global__ __launch_bounds__(128) void hd_kernel(const float* __restrict__ rowmin_partial,
                                                 const float* __restrict__ colmax_partial,
                                                 float* __restrict__ hd_out) {
    const int b = blockIdx.x;
    const int tid = threadIdx.x;  // = y
    __shared__ float sm[128];
    float cm = -2.0f;
#pragma unroll 4
    for (int xt = 0; xt < 32; ++xt)
        cm = fmaxf(cm, colmax_partial[((size_t)(b * 32 + xt)) * LY + tid]);
    sm[tid] = cm;
    __syncthreads();
    for (int s = 64; s > 0; s >>= 1) {
        if (tid < s) sm[tid] = fminf(sm[tid], sm[tid + s]);
        __syncthreads();
    }
    if (tid == 0) {
        float a = sm[0];
        float rm = 1e30f;
        for (int xt = 0; xt < 32; ++xt) rm = fminf(rm, rowmin_partial[b * 32 + xt]);
        a = fminf(a, rm);
        hd_out[b] = sqrtf(fmaxf(2.0f - 2.0f * a, 0.0f));
    }
}

// ---------------------------------------------------------------------------
// Kernel 4: MSE partial sums over 1,048,576 elements. 256 blocks x 256 thr.
// ---------------------------------------------------------------------------
__global__ __launch_bounds__(256) void mse_kernel(const float* __restrict__ P,
                                                  const float* __restrict__ M,
                                                  float* __restrict__ partial) {
    const int tid = blockIdx.x * 256 + threadIdx.x;  // 65536 threads
    float s = 0.0f;
#pragma unroll
    for (int j = 0; j < 4; ++j) {
        const int i = tid + j * 65536;  // float4 index, 262144 total
        const float4 p = *(const float4*)(P + (size_t)i * 4);
        const float4 m = *(const float4*)(M + (size_t)i * 4);
        float dx = p.x - m.x, dy = p.y - m.y, dz = p.z - m.z, dw = p.w - m.w;
        s += dx * dx + dy * dy + dz * dz + dw * dw;
    }
    __shared__ float sm[256];
    sm[threadIdx.x] = s;
    __syncthreads();
    for (int st = 128; st > 0; st >>= 1) {
        if (threadIdx.x < st) sm[threadIdx.x] += sm[threadIdx.x + st];
        __syncthreads();
    }
    if (threadIdx.x == 0) partial[blockIdx.x] = sm[0];
}

// ---------------------------------------------------------------------------
// Kernel 5: final assembly: bce + mse + contrast. One block of 256 threads.
// ---------------------------------------------------------------------------
__global__ __launch_bounds__(256) void final_kernel(const float* __restrict__ partial,
                                                    const float* __restrict__ hd,
                                                    const float* __restrict__ prob,
                                                    const int* __restrict__ label,
                                                    float* __restrict__ out) {
    const int tid = threadIdx.x;
    __shared__ float s1[256];  // mse partial sums
    __shared__ float s2[256];  // contrast + bce per-b contributions
    float msum = partial[tid];
    float cb = 0.0f;
    if (tid < BB) {
        float lab = (float)label[tid];
        float h = hd[tid];
        float r = fmaxf(0.5f - h, 0.0f);
        float contrast = lab * h * h + (1.0f - lab) * r * r;
        float p = prob[tid];
        float logp   = fmaxf(logf(p), -100.0f);
        float log1mp = fmaxf(log1pf(-p), -100.0f);
        float bce = -(lab * logp + (1.0f - lab) * log1mp);
        cb = (contrast + bce) * (1.0f / (float)BB);
    }
    s1[tid] = msum;
    s2[tid] = cb;
    __syncthreads();
    for (int st = 128; st > 0; st >>= 1) {
        if (tid < st) { s1[tid] += s1[tid + st]; s2[tid] += s2[tid + st]; }
        __syncthreads();
    }
    if (tid == 0) out[0] = s2[0] + s1[0] * (1.0f / 1048576.0f);
}

// ---------------------------------------------------------------------------
extern "C" void kernel_launch(void* const* d_in, const int* in_sizes, int n_in,
                              void* d_out, int out_size, void* d_ws, size_t ws_size,
                              hipStream_t stream) {
    const float* X    = (const float*)d_in[0];  // encoded_frame [512,64,512]
    const float* Y    = (const float*)d_in[1];  // encoded_word  [128,64,512]
    const float* Pm   = (const float*)d_in[2];  // predict_mask_feature [64,32,512]
    const float* Mm   = (const float*)d_in[3];  // masked_feature       [64,32,512]
    const float* prob = (const float*)d_in[4];  // prob_v [64,1]
    const int*   lab  = (const int*)d_in[5];    // label [64]
    float* out = (float*)d_out;

    float* ws = (float*)d_ws;
    float* rnx            = ws;                       // 32768
    float* rny            = rnx + LX * BB;            // 8192
    float* rowmin_partial = rny + LY * BB;            // 2048
    float* colmax_partial = rowmin_partial + BB * 32; // 262144
    float* mse_partial    = colmax_partial + (size_t)BB * 32 * LY;  // 256
    float* hd_ws          = mse_partial + 256;        // 64

    // 1) inverse norms: (512*64 + 128*64) rows / 8 waves per block
    norm_kernel<<<(LX * BB + LY * BB) / 8, 256, 0, stream>>>(X, Y, rnx, rny);
    // 2) WMMA GEMM + row/col max partials: 64 b * 4 x-groups
    gemm_hd_kernel<<<BB * 4, 256, 0, stream>>>(X, Y, rnx, rny,
                                               rowmin_partial, colmax_partial);
    // 3) hd[b]
    hd_kernel<<<BB, 128, 0, stream>>>(rowmin_partial, colmax_partial, hd_ws);
    // 4) MSE partials
    mse_kernel<<<256, 256, 0, stream>>>(Pm, Mm, mse_partial);
    // 5) final loss
    final_kernel<<<1, 256, 0, stream>>>(mse_partial, hd_ws, prob, lab, out);
}